// NGCF_67147518705976
// MI455X (gfx1250) — compile-verified
//
#include <hip/hip_runtime.h>

#define NUM_USERS 100000
#define NUM_ITEMS 50000
#define N_NODES   150000
#define EMB       64
#define NNZ_CNT   2400000
#define BATCH_SZ  16384
#define FINAL_LD  192   // concat(emb, layer1, layer2)

typedef __attribute__((ext_vector_type(2))) float v2f;
typedef __attribute__((ext_vector_type(8))) float v8f;

// ---------------------------------------------------------------------------
// Stage 0: FINAL[:, 0:64] = concat(uEmb, iEmb)
// ---------------------------------------------------------------------------
__global__ void k_init_features(const float* __restrict__ uEmb,
                                const float* __restrict__ iEmb,
                                float* __restrict__ FINAL) {
    size_t t = (size_t)blockIdx.x * blockDim.x + threadIdx.x;
    if (t >= (size_t)N_NODES * EMB) return;
    int r = (int)(t / EMB);
    int d = (int)(t % EMB);
    float v = (r < NUM_USERS) ? uEmb[(size_t)r * EMB + d]
                              : iEmb[(size_t)(r - NUM_USERS) * EMB + d];
    FINAL[(size_t)r * FINAL_LD + d] = v;
}

// ---------------------------------------------------------------------------
// Stage 1a: H = features (self-loop term), features = FINAL[:, colBase:colBase+64]
// ---------------------------------------------------------------------------
__global__ void k_h_init(const float* __restrict__ FINAL,
                         float* __restrict__ H, int colBase) {
    size_t t = (size_t)blockIdx.x * blockDim.x + threadIdx.x;
    if (t >= (size_t)N_NODES * EMB) return;
    int r = (int)(t / EMB);
    int d = (int)(t % EMB);
    H[t] = FINAL[(size_t)r * FINAL_LD + colBase + d];
}

// ---------------------------------------------------------------------------
// Stage 1b: H[rows[e], :] += vals[e] * FINAL[cols[e], colBase:colBase+64]
// 16 threads per nnz, float4 per thread, f32 global atomics.
// ---------------------------------------------------------------------------
__global__ void k_spmm(const int* __restrict__ rows,
                       const int* __restrict__ cols,
                       const float* __restrict__ vals,
                       const float* __restrict__ FINAL,
                       float* __restrict__ H, int colBase) {
    size_t t = (size_t)blockIdx.x * blockDim.x + threadIdx.x;
    if (t >= (size_t)NNZ_CNT * 16) return;
    int e = (int)(t >> 4);
    int j = (int)((t & 15) << 2);          // dim base: 0,4,...,60
    int r = rows[e];
    int c = cols[e];
    float v = vals[e];
    const float4 x = *(const float4*)(FINAL + (size_t)c * FINAL_LD + colBase + j);
    float* hp = H + (size_t)r * EMB + j;
    atomicAdd(hp + 0, v * x.x);
    atomicAdd(hp + 1, v * x.y);
    atomicAdd(hp + 2, v * x.z);
    atomicAdd(hp + 3, v * x.w);
}

// ---------------------------------------------------------------------------
// WMMA f32 16x16x4 tile GEMM: FINAL[:, outColBase+] = relu(H @ W + b)
// One wave per 16x16 output tile; K = 64 in 16 steps of 4.
// A 16x4 layout: lanes 0-15 hold K={0,1}, lanes 16-31 hold K={2,3}.
// B 4x16 layout: lane%16 = column, lane half selects K pair.
// ---------------------------------------------------------------------------
__global__ void k_layer_gemm(const float* __restrict__ A,     // [N_NODES, 64]
                             const float* __restrict__ W,     // [64, 64]
                             const float* __restrict__ bias,  // [64]
                             float* __restrict__ FINAL, int outColBase) {
    const int lane = threadIdx.x & 31;
    const int wave = threadIdx.x >> 5;
    const int tile = blockIdx.x * (blockDim.x >> 5) + wave;
    const int nTilesN = EMB / 16;                       // 4
    const int totalTiles = (N_NODES / 16) * nTilesN;    // 37500
    if (tile >= totalTiles) return;                     // whole-wave exit (EXEC stays full)
    const int mBase = (tile / nTilesN) * 16;
    const int nBase = (tile % nTilesN) * 16;
    const int half  = lane >> 4;
    const int lr    = lane & 15;

    const float* Arow = A + (size_t)(mBase + lr) * EMB;
    v8f acc = {};
#pragma unroll
    for (int k = 0; k < EMB; k += 4) {
        int k0 = k + 2 * half;
        v2f a, b;
        a.x = Arow[k0];
        a.y = Arow[k0 + 1];
        b.x = W[(size_t)k0 * EMB + nBase + lr];
        b.y = W[(size_t)(k0 + 1) * EMB + nBase + lr];
        acc = __builtin_amdgcn_wmma_f32_16x16x4_f32(
            false, a, false, b, (short)0, acc, false, false);
    }
    const float bb = bias[nBase + lr];
#pragma unroll
    for (int v = 0; v < 8; ++v) {
        int row = mBase + v + 8 * half;
        float o = acc[v] + bb;
        FINAL[(size_t)row * FINAL_LD + outColBase + nBase + lr] = fmaxf(o, 0.0f);
    }
}

// ---------------------------------------------------------------------------
// MLP layer 1: M1 = relu(gather(FINAL) @ W1 + b1).
// A row b is concat(FINAL[userIdx[b]], FINAL[itemIdx[b]+NUM_USERS]) (384 wide);
// gather fused into the A-fragment load (k<192 vs k>=192, uniform per K-step).
// ---------------------------------------------------------------------------
__global__ void k_mlp1(const int* __restrict__ userIdx,
                       const int* __restrict__ itemIdx,
                       const float* __restrict__ FINAL,
                       const float* __restrict__ W1,   // [384, 64]
                       const float* __restrict__ b1,   // [64]
                       float* __restrict__ M1) {       // [BATCH, 64]
    const int lane = threadIdx.x & 31;
    const int wave = threadIdx.x >> 5;
    const int tile = blockIdx.x * (blockDim.x >> 5) + wave;
    const int nTilesN = 4;
    const int totalTiles = (BATCH_SZ / 16) * nTilesN;   // 4096
    if (tile >= totalTiles) return;
    const int mBase = (tile / nTilesN) * 16;
    const int nBase = (tile % nTilesN) * 16;
    const int half  = lane >> 4;
    const int lr    = lane & 15;

    const int row = mBase + lr;
    const float* uRow = FINAL + (size_t)userIdx[row] * FINAL_LD;
    const float* iRow = FINAL + (size_t)(itemIdx[row] + NUM_USERS) * FINAL_LD;

    v8f acc = {};
#pragma unroll 8
    for (int k = 0; k < 2 * FINAL_LD; k += 4) {
        int k0 = k + 2 * half;
        const float* src = (k0 < FINAL_LD) ? (uRow + k0) : (iRow + (k0 - FINAL_LD));
        v2f a, b;
        a.x = src[0];
        a.y = src[1];
        b.x = W1[(size_t)k0 * 64 + nBase + lr];
        b.y = W1[(size_t)(k0 + 1) * 64 + nBase + lr];
        acc = __builtin_amdgcn_wmma_f32_16x16x4_f32(
            false, a, false, b, (short)0, acc, false, false);
    }
    const float bb = b1[nBase + lr];
#pragma unroll
    for (int v = 0; v < 8; ++v) {
        int r = mBase + v + 8 * half;
        M1[(size_t)r * 64 + nBase + lr] = fmaxf(acc[v] + bb, 0.0f);
    }
}

// ---------------------------------------------------------------------------
// MLP layer 2: M2 = M1 @ W2 + b2  (no ReLU). [BATCH,64]x[64,32]
// ---------------------------------------------------------------------------
__global__ void k_mlp2(const float* __restrict__ M1,
                       const float* __restrict__ W2,   // [64, 32]
                       const float* __restrict__ b2,   // [32]
                       float* __restrict__ M2) {       // [BATCH, 32]
    const int lane = threadIdx.x & 31;
    const int wave = threadIdx.x >> 5;
    const int tile = blockIdx.x * (blockDim.x >> 5) + wave;
    const int nTilesN = 2;
    const int totalTiles = (BATCH_SZ / 16) * nTilesN;   // 2048
    if (tile >= totalTiles) return;
    const int mBase = (tile / nTilesN) * 16;
    const int nBase = (tile % nTilesN) * 16;
    const int half  = lane >> 4;
    const int lr    = lane & 15;

    const float* Arow = M1 + (size_t)(mBase + lr) * 64;
    v8f acc = {};
#pragma unroll
    for (int k = 0; k < 64; k += 4) {
        int k0 = k + 2 * half;
        v2f a, b;
        a.x = Arow[k0];
        a.y = Arow[k0 + 1];
        b.x = W2[(size_t)k0 * 32 + nBase + lr];
        b.y = W2[(size_t)(k0 + 1) * 32 + nBase + lr];
        acc = __builtin_amdgcn_wmma_f32_16x16x4_f32(
            false, a, false, b, (short)0, acc, false, false);
    }
    const float bb = b2[nBase + lr];
#pragma unroll
    for (int v = 0; v < 8; ++v) {
        int r = mBase + v + 8 * half;
        M2[(size_t)r * 32 + nBase + lr] = acc[v] + bb;
    }
}

// ---------------------------------------------------------------------------
// MLP layer 3: out = M2 @ W3 + b3  ([BATCH,32]x[32,1] -> dot per row)
// ---------------------------------------------------------------------------
__global__ void k_mlp3(const float* __restrict__ M2,
                       const float* __restrict__ W3,   // [32]
                       const float* __restrict__ b3,   // [1]
                       float* __restrict__ out) {
    int b = blockIdx.x * blockDim.x + threadIdx.x;
    if (b >= BATCH_SZ) return;
    float s = b3[0];
#pragma unroll
    for (int j = 0; j < 32; ++j)
        s += M2[(size_t)b * 32 + j] * W3[j];
    out[b] = s;
}

// ---------------------------------------------------------------------------
extern "C" void kernel_launch(void* const* d_in, const int* in_sizes, int n_in,
                              void* d_out, int out_size, void* d_ws, size_t ws_size,
                              hipStream_t stream) {
    const int*   userIdx  = (const int*)d_in[0];
    const int*   itemIdx  = (const int*)d_in[1];
    const int*   lap_rows = (const int*)d_in[2];
    const int*   lap_cols = (const int*)d_in[3];
    const float* lap_vals = (const float*)d_in[4];
    const float* uEmb     = (const float*)d_in[5];
    const float* iEmb     = (const float*)d_in[6];
    const float* gW0      = (const float*)d_in[7];
    const float* gb0      = (const float*)d_in[8];
    const float* gW1      = (const float*)d_in[9];
    const float* gb1      = (const float*)d_in[10];
    const float* W1       = (const float*)d_in[11];
    const float* b1       = (const float*)d_in[12];
    const float* W2       = (const float*)d_in[13];
    const float* b2       = (const float*)d_in[14];
    const float* W3       = (const float*)d_in[15];
    const float* b3       = (const float*)d_in[16];

    // workspace layout (floats)
    float* FINAL = (float*)d_ws;                               // [N_NODES, 192]
    float* H     = FINAL + (size_t)N_NODES * FINAL_LD;         // [N_NODES, 64]
    float* M1    = H + (size_t)N_NODES * EMB;                  // [BATCH, 64]
    float* M2    = M1 + (size_t)BATCH_SZ * EMB;                // [BATCH, 32]

    const int TB = 256;  // 8 wave32 waves per block

    {   // FINAL[:, 0:64] = embeddings
        size_t n = (size_t)N_NODES * EMB;
        k_init_features<<<(unsigned)((n + TB - 1) / TB), TB, 0, stream>>>(uEmb, iEmb, FINAL);
    }

    const float* gWs[2] = {gW0, gW1};
    const float* gbs[2] = {gb0, gb1};
    for (int l = 0; l < 2; ++l) {
        size_t n = (size_t)N_NODES * EMB;
        k_h_init<<<(unsigned)((n + TB - 1) / TB), TB, 0, stream>>>(FINAL, H, l * EMB);
        size_t ne = (size_t)NNZ_CNT * 16;
        k_spmm<<<(unsigned)((ne + TB - 1) / TB), TB, 0, stream>>>(
            lap_rows, lap_cols, lap_vals, FINAL, H, l * EMB);
        int tiles = (N_NODES / 16) * (EMB / 16);
        k_layer_gemm<<<(tiles + 7) / 8, TB, 0, stream>>>(
            H, gWs[l], gbs[l], FINAL, (l + 1) * EMB);
    }

    int t1 = (BATCH_SZ / 16) * 4;
    k_mlp1<<<(t1 + 7) / 8, TB, 0, stream>>>(userIdx, itemIdx, FINAL, W1, b1, M1);
    int t2 = (BATCH_SZ / 16) * 2;
    k_mlp2<<<(t2 + 7) / 8, TB, 0, stream>>>(M1, W2, b2, M2);
    k_mlp3<<<(BATCH_SZ + TB - 1) / TB, TB, 0, stream>>>(M2, W3, b3, (float*)d_out);
}